// LearnedQueryAttention_22428319220517
// MI455X (gfx1250) — compile-verified
//
#include <hip/hip_runtime.h>
#include <math.h>

#define KDIM 1024
#define VDIM 1024
#define NHEADS 16
#define BSZ 16
#define SLEN 4096
#define HDIM 64
#define NCHUNK 16
#define NGROUP (BSZ * NHEADS)              /* 256 */
#define ROWS_PER_CHUNK (SLEN / NCHUNK)     /* 256 */
#define LN_EPS 1e-5f

typedef float v2f __attribute__((ext_vector_type(2)));
typedef float v8f __attribute__((ext_vector_type(8)));

/* ---- workspace layout (in floats) ---- */
#define WS_QN 0                                   /* 1024  normalized q        */
#define WS_PM (WS_QN + KDIM)                      /* 4096  partial max         */
#define WS_PL (WS_PM + NGROUP * NCHUNK)           /* 4096  partial sum-exp     */
#define WS_PA (WS_PL + NGROUP * NCHUNK)           /* 262144 partial acc[64]    */
#define WS_AF (WS_PA + NGROUP * NCHUNK * HDIM)    /* 16384 A_full [16,1024]    */
#define WS_Y  (WS_AF + BSZ * VDIM)                /* 16384 pre-LN Y [16,1024]  */

/* ------------------------------------------------------------------ */
/* Kernel 0: Qn = q / ||q||                                            */
/* ------------------------------------------------------------------ */
__global__ void lqa_qnorm(const float* __restrict__ q, float* __restrict__ qn) {
    __shared__ float red[8];
    int tid = threadIdx.x;                 /* 256 threads, 4 floats each */
    float4 v = ((const float4*)q)[tid];
    float s = v.x * v.x + v.y * v.y + v.z * v.z + v.w * v.w;
    #pragma unroll
    for (int m = 1; m < 32; m <<= 1) s += __shfl_xor(s, m, 32);
    if ((tid & 31) == 0) red[tid >> 5] = s;
    __syncthreads();
    float tot = 0.f;
    #pragma unroll
    for (int i = 0; i < 8; ++i) tot += red[i];
    float inv = rsqrtf(tot);
    float4 o;
    o.x = v.x * inv; o.y = v.y * inv; o.z = v.z * inv; o.w = v.w * inv;
    ((float4*)qn)[tid] = o;
}

/* ------------------------------------------------------------------ */
/* Kernel 1: streaming online-softmax attention partials.              */
/* grid = (NCHUNK, NGROUP), block = 256 (8 waves).                     */
/* Each 16-lane half-wave owns an online-softmax state; a wave reads   */
/* two adjacent rows per iteration -> 512 contiguous bytes per vmem op.*/
/* ------------------------------------------------------------------ */
__global__ void lqa_attn_partial(const float* __restrict__ Kp,
                                 const float* __restrict__ Vp,
                                 const float* __restrict__ qn,
                                 float* __restrict__ part_m,
                                 float* __restrict__ part_l,
                                 float* __restrict__ part_acc) {
    __shared__ float sm_m[16];
    __shared__ float sm_l[16];
    __shared__ float sm_acc[16][HDIM];

    const int g     = blockIdx.y;             /* group in [0,256) */
    const int chunk = blockIdx.x;             /* chunk in [0,16)  */
    const int head  = g & (NHEADS - 1);
    const int tid   = threadIdx.x;
    const int wave  = tid >> 5;
    const int lane  = tid & 31;
    const int half  = lane >> 4;              /* which 16-lane half */
    const int lh    = lane & 15;              /* lane in half       */

    const size_t base = (size_t)g * (SLEN * HDIM) +
                        (size_t)chunk * ROWS_PER_CHUNK * HDIM;
    const float* qseg = qn + head * HDIM;
    const float4 qv = *(const float4*)(qseg + 4 * lh);

    float m = -3.402823466e38f;
    float l = 0.f;
    float ax = 0.f, ay = 0.f, az = 0.f, aw = 0.f;
    const float scale = 0.125f;               /* 1/sqrt(64) */

    #pragma unroll 4
    for (int i = 0; i < 16; ++i) {
        const int row = wave * 32 + 2 * i + half;
        const float4 kv = *(const float4*)(Kp + base + (size_t)row * HDIM + 4 * lh);
        const float4 vv = *(const float4*)(Vp + base + (size_t)row * HDIM + 4 * lh);
        float s = kv.x * qv.x + kv.y * qv.y + kv.z * qv.z + kv.w * qv.w;
        /* reduce across the 16-lane half (xor masks stay inside the half) */
        s += __shfl_xor(s, 1, 32);
        s += __shfl_xor(s, 2, 32);
        s += __shfl_xor(s, 4, 32);
        s += __shfl_xor(s, 8, 32);
        s *= scale;
        const float mn   = fmaxf(m, s);
        const float corr = __expf(m - mn);
        const float p    = __expf(s - mn);
        l  = l * corr + p;
        ax = ax * corr + p * vv.x;
        ay = ay * corr + p * vv.y;
        az = az * corr + p * vv.z;
        aw = aw * corr + p * vv.w;
        m  = mn;
    }

    /* publish half-wave states */
    const int hid = wave * 2 + half;          /* 0..15 */
    if (lh == 0) { sm_m[hid] = m; sm_l[hid] = l; }
    sm_acc[hid][4 * lh + 0] = ax;
    sm_acc[hid][4 * lh + 1] = ay;
    sm_acc[hid][4 * lh + 2] = az;
    sm_acc[hid][4 * lh + 3] = aw;
    __syncthreads();

    /* merge 16 states -> one block partial; 64 owner threads */
    if (tid < HDIM) {
        float M = -3.402823466e38f;
        #pragma unroll
        for (int j = 0; j < 16; ++j) M = fmaxf(M, sm_m[j]);
        float L = 0.f, A = 0.f;
        #pragma unroll
        for (int j = 0; j < 16; ++j) {
            const float w = __expf(sm_m[j] - M);
            L += sm_l[j] * w;
            A += sm_acc[j][tid] * w;
        }
        const int pidx = g * NCHUNK + chunk;
        part_acc[(size_t)pidx * HDIM + tid] = A;
        if (tid == 0) { part_m[pidx] = M; part_l[pidx] = L; }
    }
}

/* ------------------------------------------------------------------ */
/* Kernel 2: merge chunk partials -> A_full [16,1024] (flat g*64+e).   */
/* ------------------------------------------------------------------ */
__global__ void lqa_combine(const float* __restrict__ part_m,
                            const float* __restrict__ part_l,
                            const float* __restrict__ part_acc,
                            float* __restrict__ a_full) {
    const int g = blockIdx.x;                 /* 256 blocks */
    const int e = threadIdx.x;                /* 64 threads */
    float M = -3.402823466e38f;
    #pragma unroll
    for (int j = 0; j < NCHUNK; ++j) M = fmaxf(M, part_m[g * NCHUNK + j]);
    float L = 0.f, A = 0.f;
    #pragma unroll
    for (int j = 0; j < NCHUNK; ++j) {
        const float w = __expf(part_m[g * NCHUNK + j] - M);
        L += part_l[g * NCHUNK + j] * w;
        A += part_acc[(size_t)(g * NCHUNK + j) * HDIM + e] * w;
    }
    a_full[g * HDIM + e] = A / L;
}

/* ------------------------------------------------------------------ */
/* Kernel 3: Y = A_full @ W^T + bias via V_WMMA_F32_16X16X4_F32.       */
/* 16 blocks x 4 waves; each wave computes one 16x16 output tile,      */
/* accumulating K=1024 in 256 wmma steps.                              */
/* A frag: lane<16 -> M=lane, K=k0..k0+1 ; lane>=16 -> M=lane-16,      */
/* K=k0+2..k0+3. B frag mirrors with N.                                */
/* ------------------------------------------------------------------ */
__global__ void lqa_gemm_wmma(const float* __restrict__ a_full,
                              const float* __restrict__ w0_w,
                              const float* __restrict__ w0_b,
                              float* __restrict__ y) {
    const int tid  = threadIdx.x;             /* 128 */
    const int wave = tid >> 5;
    const int lane = tid & 31;
    const int tile = blockIdx.x * 4 + wave;   /* 0..63 -> n tile */

    const int mrow = lane & 15;
    const int ncol = tile * 16 + (lane & 15);
    const int ksel = (lane >> 4) * 2;

    v8f c = {};
    #pragma unroll 8
    for (int k0 = 0; k0 < KDIM; k0 += 4) {
        v2f a = *(const v2f*)(a_full + (size_t)mrow * KDIM + k0 + ksel);
        v2f b = *(const v2f*)(w0_w  + (size_t)ncol * KDIM + k0 + ksel);
        c = __builtin_amdgcn_wmma_f32_16x16x4_f32(
                /*neg_a=*/false, a, /*neg_b=*/false, b,
                /*c_mod=*/(short)0, c, /*reuse_a=*/false, /*reuse_b=*/false);
    }

    const float bias = w0_b[ncol];
    #pragma unroll
    for (int r = 0; r < 8; ++r) {
        const int mr = (lane < 16) ? r : (r + 8);
        y[(size_t)mr * VDIM + ncol] = c[r] + bias;
    }
}

/* ------------------------------------------------------------------ */
/* Kernel 4: LayerNorm over last dim (population variance).            */
/* ------------------------------------------------------------------ */
__global__ void lqa_layernorm(const float* __restrict__ y,
                              const float* __restrict__ ln_g,
                              const float* __restrict__ ln_b,
                              float* __restrict__ out) {
    __shared__ float rs[8];
    __shared__ float rss[8];
    const int b   = blockIdx.x;               /* 16 rows */
    const int tid = threadIdx.x;              /* 256 threads, 4 cols each */
    const float4 v = ((const float4*)(y + (size_t)b * VDIM))[tid];
    float s  = v.x + v.y + v.z + v.w;
    float ss = v.x * v.x + v.y * v.y + v.z * v.z + v.w * v.w;
    #pragma unroll
    for (int m = 1; m < 32; m <<= 1) {
        s  += __shfl_xor(s,  m, 32);
        ss += __shfl_xor(ss, m, 32);
    }
    if ((tid & 31) == 0) { rs[tid >> 5] = s; rss[tid >> 5] = ss; }
    __syncthreads();
    float S = 0.f, SS = 0.f;
    #pragma unroll
    for (int i = 0; i < 8; ++i) { S += rs[i]; SS += rss[i]; }
    const float mu  = S * (1.0f / VDIM);
    const float var = SS * (1.0f / VDIM) - mu * mu;
    const float inv = rsqrtf(var + LN_EPS);

    const float4 gg = ((const float4*)ln_g)[tid];
    const float4 bb = ((const float4*)ln_b)[tid];
    float4 o;
    o.x = gg.x * (v.x - mu) * inv + bb.x;
    o.y = gg.y * (v.y - mu) * inv + bb.y;
    o.z = gg.z * (v.z - mu) * inv + bb.z;
    o.w = gg.w * (v.w - mu) * inv + bb.w;
    ((float4*)(out + (size_t)b * VDIM))[tid] = o;
}

/* ------------------------------------------------------------------ */
extern "C" void kernel_launch(void* const* d_in, const int* in_sizes, int n_in,
                              void* d_out, int out_size, void* d_ws, size_t ws_size,
                              hipStream_t stream) {
    const float* Kp   = (const float*)d_in[0];
    const float* Vp   = (const float*)d_in[1];
    const float* q    = (const float*)d_in[2];
    const float* w0w  = (const float*)d_in[3];
    const float* w0b  = (const float*)d_in[4];
    const float* lng  = (const float*)d_in[5];
    const float* lnb  = (const float*)d_in[6];
    float*       out  = (float*)d_out;

    float* ws      = (float*)d_ws;
    float* qn      = ws + WS_QN;
    float* part_m  = ws + WS_PM;
    float* part_l  = ws + WS_PL;
    float* part_a  = ws + WS_PA;
    float* a_full  = ws + WS_AF;
    float* y       = ws + WS_Y;

    lqa_qnorm<<<1, 256, 0, stream>>>(q, qn);
    lqa_attn_partial<<<dim3(NCHUNK, NGROUP), 256, 0, stream>>>(
        Kp, Vp, qn, part_m, part_l, part_a);
    lqa_combine<<<NGROUP, HDIM, 0, stream>>>(part_m, part_l, part_a, a_full);
    lqa_gemm_wmma<<<16, 128, 0, stream>>>(a_full, w0w, w0b, y);
    lqa_layernorm<<<16, 256, 0, stream>>>(y, lng, lnb, out);
}